// ContrasTR_36962488549919
// MI455X (gfx1250) — compile-verified
//
#include <hip/hip_runtime.h>
#include <stdint.h>

// ---- problem constants (from the reference) ----
#define NB 64
#define NQ 900
#define ND 256
#define NC 91
#define NM 4500            // Q * MAX_PREV_FRAMES
#define MAXPREV 5
// sigmoid(x) >= 0.8  <=>  x >= ln(0.8/0.2) = ln 4
#define LOGIT_THRESH 1.3862943611198906f

typedef float v4f __attribute__((ext_vector_type(4)));

// =====================================================================
// Block-wide stable exclusive scan of a 0/1 flag (1024 threads, wave32).
// Returns this thread's rank among flag==1 threads; *total = block sum.
// =====================================================================
__device__ __forceinline__ int block_scan_flag(int flag, int tid, int* wsum, int* total) {
    const int lane = tid & 31;
    const int wv   = tid >> 5;                       // up to 32 waves
    unsigned bal = __builtin_amdgcn_ballot_w32(flag != 0);
    int lane_prefix = __popc(bal & ((1u << lane) - 1u));
    if (lane == 0) wsum[wv] = __popc(bal);
    __syncthreads();
    if (tid == 0) {
        int acc = 0;
        for (int w = 0; w < 32; ++w) { int v = wsum[w]; wsum[w] = acc; acc += v; }
        wsum[32] = acc;
    }
    __syncthreads();
    int rank = wsum[wv] + lane_prefix;
    *total = wsum[32];
    __syncthreads();                                 // wsum reused by next call
    return rank;
}

// =====================================================================
// Kernel A: per-batch eviction/compaction map + prediction selection.
// Writes src_map (d_ws) and the scalar outputs (fn/tid/mask/nb) as f32.
//   src_map[b*NM + slot]  : -1 -> empty, [0,NM) -> kept mem row,
//                           NM+q -> selected prediction q.
// =====================================================================
__global__ __launch_bounds__(1024) void build_maps_kernel(
    const int* __restrict__ mem_fn, const int* __restrict__ mem_tid,
    const unsigned char* __restrict__ mem_mask,
    const float* __restrict__ pred_classes, const int* __restrict__ pred_ids,
    const int* __restrict__ cur_frames,
    int* __restrict__ src_map,
    float* __restrict__ out_fn, float* __restrict__ out_tid,
    float* __restrict__ out_mask, float* __restrict__ out_nb)
{
    __shared__ int wsum[33];
    const int b    = blockIdx.x;
    const int tid  = threadIdx.x;
    const int cur  = cur_frames[b];
    const int mb   = b * NM;

    // ---- stable compaction of kept memory entries ----
    int base = 0;
    for (int c0 = 0; c0 < NM; c0 += 1024) {
        const int i = c0 + tid;
        int flag = 0;
        if (i < NM)
            flag = (mem_fn[mb + i] > cur - MAXPREV) && (mem_mask[mb + i] != 0);
        int tot;
        int rank = block_scan_flag(flag, tid, wsum, &tot);
        if (flag) src_map[mb + base + rank] = i;
        base += tot;                                  // uniform across threads
    }
    const int nb_kept = base;

    // fill tail with -1 (disjoint from kept slots)
    for (int i = nb_kept + tid; i < NM; i += 1024) src_map[mb + i] = -1;
    __syncthreads();

    // ---- thresholded predictions (Q=900 fits one 1024-chunk) ----
    int flag = 0;
    if (tid < NQ) {
        const float* pc = pred_classes + ((size_t)b * NQ + tid) * NC;
        float mx = pc[0];
        #pragma unroll 7
        for (int c = 1; c < NC; ++c) mx = fmaxf(mx, pc[c]);
        flag = (mx >= LOGIT_THRESH);
    }
    int nb_pred;
    int rank = block_scan_flag(flag, tid, wsum, &nb_pred);
    const int dest = nb_kept + rank;
    if (flag && dest < NM) src_map[mb + dest] = NM + tid;
    __syncthreads();

    int nb_items = nb_kept + nb_pred;
    if (nb_items > NM) nb_items = NM;

    // ---- scalar outputs (harness output buffer is f32, streaming) ----
    for (int i = tid; i < NM; i += 1024) {
        const int s = src_map[mb + i];
        float fn = -1.0f, td = 0.0f, mk = 0.0f;
        if (s >= 0) {
            mk = 1.0f;
            if (s < NM) { fn = (float)mem_fn[mb + s]; td = (float)mem_tid[mb + s]; }
            else        { fn = (float)cur;            td = (float)pred_ids[b * NQ + (s - NM)]; }
        }
        __builtin_nontemporal_store(fn, out_fn  + mb + i);
        __builtin_nontemporal_store(td, out_tid + mb + i);
        __builtin_nontemporal_store(mk, out_mask + mb + i);
    }
    if (tid == 0) out_nb[b] = (float)nb_items;
}

// =====================================================================
// Kernel B: gather-copy of 1KB embedding rows via CDNA5 async DMA.
// One wave (32 lanes) per row; per-wave 2KB LDS double buffer.
// Pipeline: async-load(row k) || drain(row k-1); loads retire in order,
// so s_wait_asynccnt <= 2 (the two in-flight loads of row k) guarantees
// row k-1's data has landed in LDS. Output stream is written NT (read
// never again here; keeps the 192MB L2 for the gathered read stream).
// =====================================================================
__global__ __launch_bounds__(256) void gather_copy_kernel(
    const float* __restrict__ mem_emb, const float* __restrict__ pred_emb,
    const int* __restrict__ src_map, float* __restrict__ out_emb,
    int nwaves_total)
{
    __shared__ __align__(16) char smem[8 * 2048];    // only LDS object -> offset 0
    const int lane = threadIdx.x & 31;
    const int wv   = threadIdx.x >> 5;               // 8 waves / block
    const int gw   = blockIdx.x * 8 + wv;
    const unsigned lds_wave = (unsigned)(wv * 2048);
    const int ROWS = NB * NM;

    int      parity    = 0;
    int      pending   = 0;
    unsigned pend_base = 0;
    v4f*     pend_dst  = nullptr;

    for (int r = gw; r < ROWS; r += nwaves_total) {
        const int bb = r / NM;
        const int s  = src_map[r];
        v4f* orow = (v4f*)(out_emb + (size_t)r * ND);

        if (s >= 0) {
            const float* grow = (s < NM)
                ? (mem_emb  + ((size_t)bb * NM + s) * ND)
                : (pred_emb + ((size_t)bb * NQ + (s - NM)) * ND);
            const unsigned cur_base = lds_wave + (unsigned)parity * 1024u;
            const unsigned ldsv = cur_base + (unsigned)(lane * 16);
            const unsigned long long ga =
                (unsigned long long)(uintptr_t)grow + (unsigned long long)(lane * 16);
            // 2 x 512B DMA: global -> LDS, bypassing VGPRs (ASYNCcnt += 2).
            // Source rows are read exactly once -> non-temporal load hint.
            asm volatile(
                "global_load_async_to_lds_b128 %0, %1, off th:TH_LOAD_NT\n\t"
                "global_load_async_to_lds_b128 %0, %1, off offset:512 th:TH_LOAD_NT"
                :: "v"(ldsv), "v"(ga) : "memory");

            if (pending) {
                // allow only the 2 just-issued loads to remain in flight
                asm volatile("s_wait_asynccnt 0x2" ::: "memory");
                const v4f* sb = (const v4f*)&smem[pend_base];
                v4f a0 = sb[lane];
                v4f a1 = sb[lane + 32];
                __builtin_nontemporal_store(a0, pend_dst + lane);
                __builtin_nontemporal_store(a1, pend_dst + lane + 32);
            }
            pending   = 1;
            pend_base = cur_base;
            pend_dst  = orow;
            parity ^= 1;
        } else {
            // empty slot: write zeros directly (no LDS involved)
            v4f z = (v4f)0.0f;
            __builtin_nontemporal_store(z, orow + lane);
            __builtin_nontemporal_store(z, orow + lane + 32);
        }
    }

    if (pending) {
        asm volatile("s_wait_asynccnt 0x0" ::: "memory");
        const v4f* sb = (const v4f*)&smem[pend_base];
        v4f a0 = sb[lane];
        v4f a1 = sb[lane + 32];
        __builtin_nontemporal_store(a0, pend_dst + lane);
        __builtin_nontemporal_store(a1, pend_dst + lane + 32);
    }
}

// =====================================================================
// Host-side launch
// =====================================================================
extern "C" void kernel_launch(void* const* d_in, const int* in_sizes, int n_in,
                              void* d_out, int out_size, void* d_ws, size_t ws_size,
                              hipStream_t stream) {
    const float*         mem_emb    = (const float*)d_in[0];
    const int*           mem_fn     = (const int*)d_in[1];
    const int*           mem_tid    = (const int*)d_in[2];
    const unsigned char* mem_mask   = (const unsigned char*)d_in[3]; // jnp bool = 1 byte
    const float*         pred_emb   = (const float*)d_in[4];
    const float*         pred_cls   = (const float*)d_in[5];
    const int*           pred_ids   = (const int*)d_in[6];
    const int*           cur_frames = (const int*)d_in[7];

    float* out      = (float*)d_out;                  // outputs concatenated as f32
    float* out_emb  = out;                            // B*M*D
    float* out_fn   = out_emb  + (size_t)NB * NM * ND;
    float* out_tid  = out_fn   + (size_t)NB * NM;
    float* out_mask = out_tid  + (size_t)NB * NM;
    float* out_nb   = out_mask + (size_t)NB * NM;

    int* src_map = (int*)d_ws;                        // B*M ints = ~1.15 MB

    build_maps_kernel<<<NB, 1024, 0, stream>>>(
        mem_fn, mem_tid, mem_mask, pred_cls, pred_ids, cur_frames,
        src_map, out_fn, out_tid, out_mask, out_nb);

    const int blocks = 2048;                          // 16384 waves, ~18 rows each
    gather_copy_kernel<<<blocks, 256, 0, stream>>>(
        mem_emb, pred_emb, src_map, out_emb, blocks * 8);
}